// Text_mlp_45354854645960
// MI455X (gfx1250) — compile-verified
//
#include <hip/hip_runtime.h>
#include <hip/hip_bf16.h>
#include <math.h>

// Problem constants (from reference)
#define B_     32
#define S_     4096
#define DIN    512
#define DOUT   300
#define W_     1024
#define M_TOT  (B_ * W_)          // 32768 rows
#define EPS_   1e-5f

typedef float v2f  __attribute__((ext_vector_type(2)));
typedef float v8f  __attribute__((ext_vector_type(8)));

// ---------------------------------------------------------------------------
// K1: ragged span means -> ttt[b, w, 0:512]  (zero for masked words)
// One block per (b,w); 128 threads, each owns 4 consecutive channels.
// ---------------------------------------------------------------------------
__global__ __launch_bounds__(128)
void k1_span_means(const float* __restrict__ t_lhs,
                   const int*   __restrict__ starts,
                   const int*   __restrict__ ends,
                   const unsigned char* __restrict__ mask,
                   float* __restrict__ ttt)
{
    const int row = blockIdx.x;            // b*W + w
    const int b   = row >> 10;             // /1024
    const int d   = threadIdx.x * 4;       // channel base (0..508)

    float4 acc = make_float4(0.f, 0.f, 0.f, 0.f);
    if (!mask[row]) {
        const int s = starts[row];
        const int e = ends[row];
        const float inv = 1.0f / (float)max(e - s, 1);
        // tt = t_lhs[:, 1:-1, :]  => global row index is (1 + r)
        const float* base = t_lhs + ((size_t)b * S_ + 1 + s) * DIN + d;
        for (int r = s; r < e; ++r) {
            float4 v = *(const float4*)base;
            acc.x += v.x; acc.y += v.y; acc.z += v.z; acc.w += v.w;
            base += DIN;
        }
        acc.x *= inv; acc.y *= inv; acc.z *= inv; acc.w *= inv;
    }
    *(float4*)(ttt + (size_t)row * DIN + d) = acc;
}

// ---------------------------------------------------------------------------
// K2: x = ttt @ W_lin^T + b_lin  via V_WMMA_F32_16X16X4_F32
// grid = (M_TOT/128, ceil(DOUT/16)); block = 256 (8 waves).
// Block stages a 16-col slice of W_lin (K-major per output col) in LDS.
// Each wave computes one 16x16 tile, looping K=512 in steps of 4.
// ---------------------------------------------------------------------------
#define BPITCH 516   // 512 + 4 floats pad: conflict-free b64 LDS reads, 16B aligned rows

__global__ __launch_bounds__(256)
void k2_gemm_wmma(const float* __restrict__ ttt,
                  const float* __restrict__ W_lin,
                  const float* __restrict__ b_lin,
                  float* __restrict__ xout)
{
    __shared__ float Bs[16 * BPITCH];      // Bs[n_local * BPITCH + k] = W_lin[n0+n][k]

    const int tid   = threadIdx.x;
    const int n0    = blockIdx.y * 16;
    const int m0b   = blockIdx.x * 128;

    // Cooperative load of 16 rows x 512 K of W_lin into LDS (float4, guarded).
    {
        const int nrow = tid >> 4;         // 0..15
        const int kl   = (tid & 15) * 4;   // 0..60
        const bool ok  = (n0 + nrow) < DOUT;
        const float* src = W_lin + (size_t)(n0 + nrow) * DIN;
        #pragma unroll
        for (int i = 0; i < 8; ++i) {
            const int k = kl + i * 64;
            float4 v = ok ? *(const float4*)(src + k)
                          : make_float4(0.f, 0.f, 0.f, 0.f);
            *(float4*)(&Bs[nrow * BPITCH + k]) = v;
        }
    }
    __syncthreads();

    const int wave = tid >> 5;             // 0..7
    const int lane = tid & 31;
    const int m0   = m0b + wave * 16;

    // ISA fragment lane mapping (wave32):
    //   A 16x4 f32: lanes 0-15 -> M=lane, K={k,k+1}; lanes 16-31 -> M=lane-16, K={k+2,k+3}
    //   B 4x16 f32: mirrored with N in place of M.
    const int frow = lane & 15;            // M (A) / N (B) within tile
    const int koff = (lane >> 4) << 1;     // 0 or 2

    const float* aptr = ttt + (size_t)(m0 + frow) * DIN + koff;

    v8f c = {};
    #pragma unroll 4
    for (int k = 0; k < DIN; k += 4) {
        v2f a = *(const v2f*)(aptr + k);
        v2f bfrag = *(const v2f*)(&Bs[frow * BPITCH + k + koff]);
        c = __builtin_amdgcn_wmma_f32_16x16x4_f32(
                /*neg_a=*/false, a, /*neg_b=*/false, bfrag,
                /*c_mod=*/(short)0, c, /*reuse_a=*/false, /*reuse_b=*/false);
    }

    // Epilogue: C/D layout -> VGPR r: lanes 0-15 M=r, lanes 16-31 M=r+8; N=lane&15
    const int n = n0 + frow;
    if (n < DOUT) {
        const float bias = b_lin[n];
        const int mbase = m0 + ((lane >> 4) << 3);
        #pragma unroll
        for (int r = 0; r < 8; ++r) {
            xout[(size_t)(mbase + r) * DOUT + n] = c[r] + bias;
        }
    }
}

// ---------------------------------------------------------------------------
// K3: per-row LayerNorm(300) -> text_output (in place), then score -> ws
// One wave per row; 8 rows per 256-thread block.
// ---------------------------------------------------------------------------
__global__ __launch_bounds__(256)
void k3_layernorm_score(float* __restrict__ x,          // in: x, out: text_output
                        const float* __restrict__ gamma,
                        const float* __restrict__ beta,
                        const float* __restrict__ w_score,
                        const float* __restrict__ b_score,
                        const unsigned char* __restrict__ mask,
                        const int* __restrict__ lam,
                        float* __restrict__ scores)
{
    const int wave = threadIdx.x >> 5;
    const int lane = threadIdx.x & 31;
    const int m    = blockIdx.x * 8 + wave;

    float* row = x + (size_t)m * DOUT;

    float v[10];
    float s = 0.f, ss = 0.f;
    #pragma unroll
    for (int j = 0; j < 10; ++j) {
        const int n = lane + 32 * j;
        float t = (n < DOUT) ? row[n] : 0.f;
        v[j] = t;
        s  += t;
        ss += t * t;
    }
    // wave32 butterfly reduction
    #pragma unroll
    for (int off = 16; off > 0; off >>= 1) {
        s  += __shfl_xor(s,  off, 32);
        ss += __shfl_xor(ss, off, 32);
    }
    const float mu   = s * (1.0f / DOUT);
    const float var  = ss * (1.0f / DOUT) - mu * mu;
    const float rstd = rsqrtf(var + EPS_);

    float dot = 0.f;
    #pragma unroll
    for (int j = 0; j < 10; ++j) {
        const int n = lane + 32 * j;
        if (n < DOUT) {
            float y = (v[j] - mu) * rstd * gamma[n] + beta[n];
            row[n] = y;
            dot += y * w_score[n];
        }
    }
    #pragma unroll
    for (int off = 16; off > 0; off >>= 1)
        dot += __shfl_xor(dot, off, 32);

    if (lane == 0) {
        float sc = dot + b_score[0];
        if (mask[m]) sc = -INFINITY;
        scores[m] = sc * (float)lam[0];
    }
}

// ---------------------------------------------------------------------------
// K4: softmax over W=1024 words per batch. One 1024-thread block per batch.
// ---------------------------------------------------------------------------
__global__ __launch_bounds__(1024)
void k4_softmax(const float* __restrict__ scores, float* __restrict__ probs)
{
    __shared__ float red[1024];
    const int tid = threadIdx.x;
    const int m   = blockIdx.x * W_ + tid;

    const float sc = scores[m];
    red[tid] = sc;
    __syncthreads();
    for (int s = 512; s > 0; s >>= 1) {
        if (tid < s) red[tid] = fmaxf(red[tid], red[tid + s]);
        __syncthreads();
    }
    const float mx = red[0];
    __syncthreads();

    const float e = __expf(sc - mx);   // exp(-inf)=0 handles masked words
    red[tid] = e;
    __syncthreads();
    for (int s = 512; s > 0; s >>= 1) {
        if (tid < s) red[tid] += red[tid + s];
        __syncthreads();
    }
    probs[m] = e / red[0];
}

// ---------------------------------------------------------------------------
// K5: text_score = broadcast probs over DOUT. One block per row.
// ---------------------------------------------------------------------------
__global__ __launch_bounds__(128)
void k5_broadcast(const float* __restrict__ probs, float* __restrict__ out2)
{
    const int m = blockIdx.x;
    const float p = probs[m];
    for (int n = threadIdx.x; n < DOUT; n += 128)
        out2[(size_t)m * DOUT + n] = p;
}

// ---------------------------------------------------------------------------
extern "C" void kernel_launch(void* const* d_in, const int* in_sizes, int n_in,
                              void* d_out, int out_size, void* d_ws, size_t ws_size,
                              hipStream_t stream)
{
    const float* t_lhs   = (const float*)d_in[0];
    const float* W_lin   = (const float*)d_in[1];
    const float* b_lin   = (const float*)d_in[2];
    const float* gamma   = (const float*)d_in[3];
    const float* beta    = (const float*)d_in[4];
    const float* w_score = (const float*)d_in[5];
    const float* b_score = (const float*)d_in[6];
    const int*   starts  = (const int*)d_in[7];
    const int*   ends    = (const int*)d_in[8];
    const unsigned char* mask = (const unsigned char*)d_in[9];
    const int*   lam     = (const int*)d_in[10];

    float* out  = (float*)d_out;
    float* out1 = out;                           // text_output (M_TOT x 300)
    float* out2 = out + (size_t)M_TOT * DOUT;    // text_score  (M_TOT x 300)

    float* ws     = (float*)d_ws;
    float* ttt    = ws;                          // M_TOT x 512
    float* scores = ws + (size_t)M_TOT * DIN;    // M_TOT
    float* probs  = scores + M_TOT;              // M_TOT

    // 1) span means
    k1_span_means<<<M_TOT, 128, 0, stream>>>(t_lhs, starts, ends, mask, ttt);

    // 2) WMMA GEMM: x + bias into out1 (scratch reuse; LN pass rewrites in place)
    dim3 g2(M_TOT / 128, (DOUT + 15) / 16);      // 256 x 19
    k2_gemm_wmma<<<g2, 256, 0, stream>>>(ttt, W_lin, b_lin, out1);

    // 3) LayerNorm + score
    k3_layernorm_score<<<M_TOT / 8, 256, 0, stream>>>(out1, gamma, beta,
                                                      w_score, b_score,
                                                      mask, lam, scores);

    // 4) per-batch softmax
    k4_softmax<<<B_, 1024, 0, stream>>>(scores, probs);

    // 5) broadcast probs -> text_score
    k5_broadcast<<<M_TOT, 128, 0, stream>>>(probs, out2);
}